// MatterSimModel_37245956391122
// MI455X (gfx1250) — compile-verified
//
#include <hip/hip_runtime.h>

typedef float v2f __attribute__((ext_vector_type(2)));
typedef float v8f __attribute__((ext_vector_type(8)));

#define PAD 64            // padded per-atom bond-list stride
#define SCAN_THREADS 1024

// ---------------- init: zero counters ----------------
__global__ void ms_init_kernel(int* __restrict__ counts, int na,
                               int* __restrict__ nsAccum, int ns) {
  int i = blockIdx.x * blockDim.x + threadIdx.x;
  if (i < na) counts[i] = 0;
  if (i < ns) nsAccum[i] = 0;
}

// ---------------- tiny: structure prefix over n_atoms ----------------
__global__ void ms_prefix_kernel(const int* __restrict__ n_atoms,
                                 int* __restrict__ structPrefix, int ns) {
  if (threadIdx.x == 0 && blockIdx.x == 0) {
    int acc = 0;
    structPrefix[0] = 0;
    for (int s = 0; s < ns; ++s) { acc += n_atoms[s]; structPrefix[s + 1] = acc; }
  }
}

// ---------------- pass 1: filter + histogram + scatter into padded lists ----------------
__global__ void ms_count_scatter_kernel(const int* __restrict__ bai,
                                        const float* __restrict__ blen,
                                        const float* __restrict__ cut,
                                        int* __restrict__ counts,
                                        int* __restrict__ lists, int nb) {
  int b = blockIdx.x * blockDim.x + threadIdx.x;
  if (b >= nb) return;
  if (blen[b] <= *cut) {
    int a = bai[2 * b];
    int slot = atomicAdd(&counts[a], 1);
    if (slot < PAD) lists[(size_t)a * PAD + slot] = b;
  }
}

// ---------------- pass 2: per-atom sort (restores stable original order), counts out ----
__global__ void ms_atom_kernel(const int* __restrict__ counts,
                               int* __restrict__ lists,
                               int* __restrict__ pairVal,
                               const int* __restrict__ structPrefix, int ns,
                               int* __restrict__ nsAccum,
                               float* __restrict__ out_nti, int na) {
  int a = blockIdx.x * blockDim.x + threadIdx.x;
  if (a >= na) return;
  int c  = counts[a];
  int cl = c > PAD ? PAD : c;
  int v[PAD];
  int* lp = lists + (size_t)a * PAD;
  for (int i = 0; i < cl; ++i) v[i] = lp[i];
  for (int i = 1; i < cl; ++i) {            // insertion sort ascending (bond ids)
    int key = v[i]; int j = i - 1;
    while (j >= 0 && v[j] > key) { v[j + 1] = v[j]; --j; }
    v[j + 1] = key;
  }
  for (int i = 0; i < cl; ++i) lp[i] = v[i];
  int pair = c * (c - 1);
  pairVal[a]  = pair;
  out_nti[a]  = (float)pair;
  int s = 0;
  while (s + 1 < ns && a >= structPrefix[s + 1]) ++s;
  atomicAdd(&nsAccum[s], pair);
}

// ---------------- pass 3: exclusive scan of pairVal via WMMA (scan-as-matmul) ----------
// Inclusive 2D scan of a 16x16 row-major tile X:
//   P = L_strict * W + X * U    (W[k][*] = rowsum(k), U upper-tri incl diag)
// done as 8 chained V_WMMA_F32_16X16X4_F32 accumulations per 256-element tile.
__global__ __launch_bounds__(SCAN_THREADS)
void ms_scan_kernel(const int* __restrict__ pairVal,
                    float* __restrict__ inclTmp,
                    int* __restrict__ trip_off, int na) {
  __shared__ float tile[32][272];   // 256 tile values + 16 row sums per wave
  __shared__ float s_tot[256];
  __shared__ int   s_off[256];
  const int tid  = threadIdx.x;
  const int wave = tid >> 5;
  const int lane = tid & 31;
  const int m    = lane & 15;       // A row / B col
  const int hi   = lane >> 4;
  const int o0   = hi * 2;          // K sub-index per documented 16x4 f32 A layout
  const int ntiles = (na + 255) >> 8;
  float* T = tile[wave];

  for (int t = wave; t < ntiles; t += 32) {
    const int tbase = t << 8;
    const bool full = (tbase + 256) <= na;
#pragma unroll
    for (int k = 0; k < 8; ++k) {
      int idx = lane + k * 32;
      int g   = tbase + idx;
      T[idx] = (full || g < na) ? (float)pairVal[g] : 0.0f;
    }
    __builtin_amdgcn_wave_barrier();
    if (lane < 16) {
      float s = 0.0f;
#pragma unroll
      for (int c2 = 0; c2 < 16; ++c2) s += T[lane * 16 + c2];
      T[256 + lane] = s;
    }
    __builtin_amdgcn_wave_barrier();

    v8f C = {0.f, 0.f, 0.f, 0.f, 0.f, 0.f, 0.f, 0.f};
#pragma unroll
    for (int k = 0; k < 4; ++k) {   // C += X * U  (row-wise prefix)
      int kb = k * 4;
      v2f a, b;
      a.x = T[m * 16 + kb + o0];
      a.y = T[m * 16 + kb + o0 + 1];
      b.x = (kb + o0     <= m) ? 1.0f : 0.0f;
      b.y = (kb + o0 + 1 <= m) ? 1.0f : 0.0f;
      C = __builtin_amdgcn_wmma_f32_16x16x4_f32(false, a, false, b, (short)0, C,
                                                false, false);
    }
#pragma unroll
    for (int k = 0; k < 4; ++k) {   // C += L_strict * W  (carry-in of earlier rows)
      int kb = k * 4;
      v2f a, b;
      a.x = (kb + o0     < m) ? 1.0f : 0.0f;
      a.y = (kb + o0 + 1 < m) ? 1.0f : 0.0f;
      b.x = T[256 + kb + o0];
      b.y = T[256 + kb + o0 + 1];
      C = __builtin_amdgcn_wmma_f32_16x16x4_f32(false, a, false, b, (short)0, C,
                                                false, false);
    }
    // D layout: VGPR i = rows i / i+8, col = lane&15
    if (full) {                     // fast path: 8 unconditional stores
#pragma unroll
      for (int i = 0; i < 8; ++i) {
        int r = i + hi * 8;
        inclTmp[tbase + r * 16 + m] = C[i];
      }
    } else {
#pragma unroll
      for (int i = 0; i < 8; ++i) {
        int r = i + hi * 8;
        int g = tbase + r * 16 + m;
        if (g < na) inclTmp[g] = C[i];
      }
    }
    if (lane == 0 && t < 256) {
      float s = 0.0f;
#pragma unroll
      for (int r = 0; r < 16; ++r) s += T[256 + r];
      s_tot[t] = s;
    }
  }
  __threadfence_block();
  __syncthreads();
  if (tid == 0) {                   // tiny serial scan of tile totals (<= 79 tiles)
    int acc = 0;
    for (int t = 0; t < ntiles && t < 256; ++t) { s_off[t] = acc; acc += (int)s_tot[t]; }
  }
  __syncthreads();
  for (int g = tid; g < na; g += SCAN_THREADS) {
    trip_off[g] = (int)inclTmp[g] + s_off[(g >> 8) & 255] - pairVal[g]; // exclusive
  }
}

// ---------------- n_triple_ij per bond ----------------
__global__ void ms_ntij_kernel(const int* __restrict__ bai,
                               const float* __restrict__ blen,
                               const float* __restrict__ cut,
                               const int* __restrict__ counts,
                               float* __restrict__ out_ntij, int nb) {
  int b = blockIdx.x * blockDim.x + threadIdx.x;
  if (b >= nb) return;
  float o = 0.0f;
  if (blen[b] <= *cut) {
    int t = counts[bai[2 * b]] - 1;
    o = t > 0 ? (float)t : 0.0f;
  }
  out_ntij[b] = o;
}

// ---------------- n_triple_s emit ----------------
__global__ void ms_ns_emit_kernel(const int* __restrict__ nsAccum,
                                  float* __restrict__ out_nts, int ns) {
  int s = blockIdx.x * blockDim.x + threadIdx.x;
  if (s < ns) out_nts[s] = (float)nsAccum[s];
}

// ---------------- pass 4: pair enumeration, one wave32 per atom -------------------
// Broadcast the atom's sorted bond list into LDS with gfx1250 async-to-LDS loads,
// emit each (i,j) row as a single 64-bit packed store.
__global__ void ms_pairs_kernel(const int* __restrict__ counts,
                                const int* __restrict__ lists,
                                const int* __restrict__ trip_off,
                                float* __restrict__ out_bi,
                                int na, long long n_triple) {
  __shared__ int slist[PAD];
  int a = blockIdx.x;
  if (a >= na) return;
  int lane = threadIdx.x;
  int c  = counts[a];
  int cl = c > PAD ? PAD : c;

  for (int idx = lane; idx < cl; idx += 32) {
    unsigned long long ga =
        (unsigned long long)(const void*)(lists + (size_t)a * PAD + idx);
    unsigned lo = (unsigned)(unsigned long long)(const void*)&slist[idx];
    asm volatile("global_load_async_to_lds_b32 %0, %1, off"
                 :: "v"(lo), "v"(ga) : "memory");
  }
  asm volatile("s_wait_asynccnt 0" ::: "memory");
  __syncthreads();

  if (cl < 2) return;
  int P = cl * (cl - 1) / 2;
  long long base = (long long)trip_off[a];
  // all rows of this atom in range?  (true except in clamp/degenerate cases)
  bool ok = (base >= 0) && (base + 2ll * P <= n_triple);
  float2* __restrict__ rows = (float2*)out_bi;
  for (int p = lane; p < P; p += 32) {
    int i = 0, rem = p;
    while (rem >= cl - 1 - i) { rem -= cl - 1 - i; ++i; }
    int j = i + 1 + rem;
    float bi = (float)slist[i];
    float bj = (float)slist[j];
    long long r0 = base + p;
    long long r1 = base + P + p;
    if (ok) {
      rows[r0] = make_float2(bi, bj);
      rows[r1] = make_float2(bj, bi);
    } else {
      if (r0 >= 0 && r0 < n_triple) rows[r0] = make_float2(bi, bj);
      if (r1 >= 0 && r1 < n_triple) rows[r1] = make_float2(bj, bi);
    }
  }
}

extern "C" void kernel_launch(void* const* d_in, const int* in_sizes, int n_in,
                              void* d_out, int out_size, void* d_ws, size_t ws_size,
                              hipStream_t stream) {
  const int*   bai     = (const int*)d_in[0];
  const float* blen    = (const float*)d_in[1];
  const int*   n_atoms = (const int*)d_in[2];
  const float* cut     = (const float*)d_in[4];
  const int nb = in_sizes[1];
  const int ns = in_sizes[2];
  const int na = in_sizes[3];
  const long long n_triple =
      ((long long)out_size - (long long)nb - (long long)na - (long long)ns) / 2;

  float* out      = (float*)d_out;
  float* out_bi   = out;
  float* out_ntij = out + 2 * n_triple;
  float* out_nti  = out_ntij + nb;
  float* out_nts  = out_nti + na;

  int* ws           = (int*)d_ws;
  int* counts       = ws;
  int* lists        = counts + na;
  int* pairVal      = lists + (size_t)PAD * na;
  int* trip_off     = pairVal + na;
  float* inclTmp    = (float*)(trip_off + na);
  int* nsAccum      = (int*)(inclTmp + na);
  int* structPrefix = nsAccum + ns;

  int initN = na > ns ? na : ns;
  ms_init_kernel<<<(initN + 255) / 256, 256, 0, stream>>>(counts, na, nsAccum, ns);
  ms_prefix_kernel<<<1, 1, 0, stream>>>(n_atoms, structPrefix, ns);
  ms_count_scatter_kernel<<<(nb + 255) / 256, 256, 0, stream>>>(
      bai, blen, cut, counts, lists, nb);
  ms_atom_kernel<<<(na + 63) / 64, 64, 0, stream>>>(
      counts, lists, pairVal, structPrefix, ns, nsAccum, out_nti, na);
  ms_scan_kernel<<<1, SCAN_THREADS, 0, stream>>>(pairVal, inclTmp, trip_off, na);
  ms_ntij_kernel<<<(nb + 255) / 256, 256, 0, stream>>>(
      bai, blen, cut, counts, out_ntij, nb);
  ms_ns_emit_kernel<<<(ns + 255) / 256, 256, 0, stream>>>(nsAccum, out_nts, ns);
  ms_pairs_kernel<<<na, 32, 0, stream>>>(counts, lists, trip_off, out_bi, na, n_triple);
}